// SimpleGATEncoder_75161927680537
// MI455X (gfx1250) — compile-verified
//
#include <hip/hip_runtime.h>
#include <hip/hip_bf16.h>

#define NODES   100000
#define NEDGES  1600000
#define DIM     128
#define EDIM    16
#define HEADS   4
#define NLAYERS 3

typedef __attribute__((ext_vector_type(2))) float v2f;
typedef __attribute__((ext_vector_type(8))) float v8f;

// ---------- order-preserving float <-> uint encoding for atomic max ----------
__device__ __forceinline__ unsigned fenc(float f) {
    unsigned u = __float_as_uint(f);
    return (u & 0x80000000u) ? ~u : (u | 0x80000000u);
}
__device__ __forceinline__ float fdec(unsigned u) {
    unsigned v = (u & 0x80000000u) ? (u & 0x7FFFFFFFu) : ~u;
    return __uint_as_float(v);
}

// =====================================================================
// Node GEMM: out[n][c] = A[n][:] @ W[:][c] + bias[c]    (A: nrows x 128)
// fp32 WMMA 16x16x4; one wave per 16-row tile, 8 N-tiles (full 128 cols).
// W staged in LDS pre-paired for the WMMA B fragment:
//   sW[r*256 + 2*col + p] = { W[4r+2p][col], W[4r+2p+1][col] },  r = k/4, p = half
// -> each B operand is ONE ds_load_b64 into an adjacent VGPR pair, the 8
//    N-tiles are constant 256 B immediate offsets, and the two half-waves
//    hit disjoint bank sets (conflict-free).
// =====================================================================
__global__ __launch_bounds__(256) void gemm_node_kernel(
    const float* __restrict__ A, const float* __restrict__ W,
    const float* __restrict__ bias, float* __restrict__ out, int nrows)
{
    __shared__ float2 sW[64 * DIM];                   // 64 KB
    // stage + swizzle: task i = (r, c4): rows 4r..4r+3, cols c4*4..c4*4+3
    for (int i = threadIdx.x; i < 32 * 32; i += 256) {
        const int r  = i >> 5;                        // 0..31
        const int c0 = (i & 31) << 2;                 // 0,4,...,124
        const float4 q0 = *(const float4*)(W + (4 * r + 0) * DIM + c0);
        const float4 q1 = *(const float4*)(W + (4 * r + 1) * DIM + c0);
        const float4 q2 = *(const float4*)(W + (4 * r + 2) * DIM + c0);
        const float4 q3 = *(const float4*)(W + (4 * r + 3) * DIM + c0);
        float2* row = sW + r * 2 * DIM;
        row[2 * (c0 + 0) + 0] = make_float2(q0.x, q1.x);
        row[2 * (c0 + 0) + 1] = make_float2(q2.x, q3.x);
        row[2 * (c0 + 1) + 0] = make_float2(q0.y, q1.y);
        row[2 * (c0 + 1) + 1] = make_float2(q2.y, q3.y);
        row[2 * (c0 + 2) + 0] = make_float2(q0.z, q1.z);
        row[2 * (c0 + 2) + 1] = make_float2(q2.z, q3.z);
        row[2 * (c0 + 3) + 0] = make_float2(q0.w, q1.w);
        row[2 * (c0 + 3) + 1] = make_float2(q2.w, q3.w);
    }
    __syncthreads();

    const int wave = threadIdx.x >> 5;
    const int lane = threadIdx.x & 31;
    const int half = lane >> 4;                       // 0: K pair (0,1); 1: K pair (2,3)
    const int lm   = lane & 15;
    const int ntiles = nrows >> 4;                    // 100000/16 = 6250 exactly
    const float2* bbase = sW + 2 * lm + half;         // per-lane base (float2 units)

    for (int t = blockIdx.x * 8 + wave; t < ntiles; t += gridDim.x * 8) {
        const int m0 = t << 4;
        v8f acc[8];
        #pragma unroll
        for (int i = 0; i < 8; i++) acc[i] = (v8f){0,0,0,0,0,0,0,0};

        const float* arow = A + (size_t)(m0 + lm) * DIM;
        for (int k = 0; k < DIM; k += 4) {
            v2f a;                                    // A 16x4 layout: v0={K0|K2}, v1={K1|K3}
            a.x = arow[k + 2 * half];
            a.y = arow[k + 2 * half + 1];
            const float2* bp = bbase + (k >> 2) * (2 * DIM);
            #pragma unroll
            for (int nt = 0; nt < 8; nt++) {
                const float2 bb = bp[nt * 32];        // constant 256 B DS offsets
                v2f b; b.x = bb.x; b.y = bb.y;
                acc[nt] = __builtin_amdgcn_wmma_f32_16x16x4_f32(
                    false, a, false, b, (short)0, acc[nt], false, false);
            }
        }
        #pragma unroll
        for (int nt = 0; nt < 8; nt++) {
            const int col = nt * 16 + lm;
            const float bv = bias[col];
            #pragma unroll
            for (int v = 0; v < 8; v++) {             // C/D: vgpr v -> row M = v + 8*half
                const int row = m0 + v + 8 * half;
                out[(size_t)row * DIM + col] = acc[nt][v] + bv;
            }
        }
    }
}

// =====================================================================
// Init: zero accumulator (aliases h), denominators, stats; maxv=0 (== -inf code)
// =====================================================================
__global__ __launch_bounds__(256) void init_kernel(
    float* acc, float* denom, unsigned* maxv, float* stats, int nrows)
{
    const size_t idx = (size_t)blockIdx.x * 256 + threadIdx.x;
    if (idx < (size_t)nrows * DIM) acc[idx] = 0.0f;
    if (idx < (size_t)nrows * HEADS) { denom[idx] = 0.0f; maxv[idx] = 0u; }
    if (idx < 2 * DIM) stats[idx] = 0.0f;
}

// =====================================================================
// Edge pass 1: e = gl[src] + gr[dst] + edge_attr@We ; leaky(0.2);
// per-head dot with att -> logits; atomic segment-max per (dst,head).
// One wave per edge; lane owns channels [4*lane, 4*lane+4); head = lane>>3.
// =====================================================================
__global__ __launch_bounds__(256) void edge_pass1_kernel(
    const float* __restrict__ gl, const float* __restrict__ gr,
    const float* __restrict__ ea, const float* __restrict__ We,
    const float* __restrict__ att,
    const long long* __restrict__ src, const long long* __restrict__ dst,
    float* __restrict__ logits, unsigned* __restrict__ maxv, int nedges)
{
    __shared__ float sWe[EDIM * DIM];                 // 8 KB, fused edge transform
    __shared__ float sAtt[DIM];
    for (int i = threadIdx.x; i < EDIM * DIM; i += 256) sWe[i] = We[i];
    if (threadIdx.x < DIM) sAtt[threadIdx.x] = att[threadIdx.x];
    __syncthreads();

    const int wave = threadIdx.x >> 5;
    const int lane = threadIdx.x & 31;
    const int e = blockIdx.x * 8 + wave;
    if (e >= nedges) return;

    const long long s = src[e];
    const long long d = dst[e];
    const float* glr = gl + (size_t)s * DIM;
    const float* grr = gr + (size_t)d * DIM;

    float eav[EDIM];                                  // one cacheline, wave-broadcast
    #pragma unroll
    for (int k = 0; k < EDIM; k++) eav[k] = ea[(size_t)e * EDIM + k];

    const int c0   = lane * 4;
    const int head = lane >> 3;
    float partial = 0.0f;
    #pragma unroll
    for (int j = 0; j < 4; j++) {
        const int c = c0 + j;
        float v = glr[c] + grr[c];
        float ee = 0.0f;
        #pragma unroll
        for (int k = 0; k < EDIM; k++) ee = fmaf(eav[k], sWe[k * DIM + c], ee);
        v += ee;
        v = v > 0.0f ? v : 0.2f * v;                  // GATv2 leaky before att-dot
        partial = fmaf(v, sAtt[c], partial);          // att flat[c] == att[head][c%32]
    }
    // reduce across the 8 lanes of this head
    partial += __shfl_xor(partial, 1);
    partial += __shfl_xor(partial, 2);
    partial += __shfl_xor(partial, 4);
    if ((lane & 7) == 0) {
        logits[(size_t)e * HEADS + head] = partial;
        atomicMax(&maxv[(size_t)d * HEADS + head], fenc(partial));
    }
}

// =====================================================================
// Edge pass 2: w = exp(logit - max[dst]); atomic denom += w;
// atomic acc[dst][c] += gl[src][c] * w   (normalized later)
// =====================================================================
__global__ __launch_bounds__(256) void edge_pass2_kernel(
    const float* __restrict__ gl, const float* __restrict__ logits,
    const unsigned* __restrict__ maxv,
    const long long* __restrict__ src, const long long* __restrict__ dst,
    float* __restrict__ denom, float* __restrict__ acc, int nedges)
{
    const int wave = threadIdx.x >> 5;
    const int lane = threadIdx.x & 31;
    const int e = blockIdx.x * 8 + wave;
    if (e >= nedges) return;

    const long long s = src[e];
    const long long d = dst[e];
    const int head = lane >> 3;

    const float lg = logits[(size_t)e * HEADS + head];
    const float m  = fdec(maxv[(size_t)d * HEADS + head]);
    const float w  = __expf(lg - m);

    if ((lane & 7) == 0) atomicAdd(&denom[(size_t)d * HEADS + head], w);

    const float* glr = gl + (size_t)s * DIM;
    float* ar = acc + (size_t)d * DIM;
    const int c0 = lane * 4;
    #pragma unroll
    for (int j = 0; j < 4; j++)
        atomicAdd(&ar[c0 + j], glr[c0 + j] * w);
}

// =====================================================================
// Finalize: acc = acc/(denom+1e-16) + bias (in place) + fused BN-stats partials
// blockDim = 128 (thread == channel), each block handles 128 rows.
// =====================================================================
__global__ __launch_bounds__(128) void finalize_kernel(
    float* __restrict__ acc, const float* __restrict__ denom,
    const float* __restrict__ bias, float* __restrict__ stats, int nrows)
{
    const int c = threadIdx.x;
    const int head = c >> 5;
    const float bv = bias[c];
    float ssum = 0.0f, ssq = 0.0f;
    const int r0 = blockIdx.x * 128;
    const int r1 = (r0 + 128 < nrows) ? (r0 + 128) : nrows;
    for (int n = r0; n < r1; n++) {
        const float dn = denom[(size_t)n * HEADS + head] + 1e-16f;
        float v = acc[(size_t)n * DIM + c] / dn + bv;
        acc[(size_t)n * DIM + c] = v;
        ssum += v;
        ssq = fmaf(v, v, ssq);
    }
    atomicAdd(&stats[c], ssum);
    atomicAdd(&stats[DIM + c], ssq);
}

// =====================================================================
// BN apply + leaky(0.02), in place (acc becomes next layer's h)
// =====================================================================
__global__ __launch_bounds__(128) void bn_apply_kernel(
    float* __restrict__ acc, const float* __restrict__ stats,
    const float* __restrict__ gamma, const float* __restrict__ beta, int nrows)
{
    const int c = threadIdx.x;
    const float inv_n = 1.0f / (float)nrows;
    const float mean  = stats[c] * inv_n;
    const float var   = stats[DIM + c] * inv_n - mean * mean;
    const float scale = rsqrtf(var + 1e-5f) * gamma[c];
    const float shift = beta[c] - mean * scale;
    const int r0 = blockIdx.x * 128;
    const int r1 = (r0 + 128 < nrows) ? (r0 + 128) : nrows;
    for (int n = r0; n < r1; n++) {
        float v = fmaf(acc[(size_t)n * DIM + c], scale, shift);
        v = v > 0.0f ? v : 0.02f * v;
        acc[(size_t)n * DIM + c] = v;
    }
}

// =====================================================================
extern "C" void kernel_launch(void* const* d_in, const int* in_sizes, int n_in,
                              void* d_out, int out_size, void* d_ws, size_t ws_size,
                              hipStream_t stream) {
    const float*     x     = (const float*)d_in[0];
    const long long* eidx  = (const long long*)d_in[1];   // int64 per reference
    const float*     ea    = (const float*)d_in[2];
    const float*     Wl    = (const float*)d_in[3];
    const float*     bl    = (const float*)d_in[4];
    const float*     Wr    = (const float*)d_in[5];
    const float*     br    = (const float*)d_in[6];
    const float*     We    = (const float*)d_in[7];
    const float*     att   = (const float*)d_in[8];
    const float*     bias  = (const float*)d_in[9];
    const float*     gamma = (const float*)d_in[10];
    const float*     beta  = (const float*)d_in[11];
    const float*     Wf    = (const float*)d_in[12];
    const float*     bf    = (const float*)d_in[13];
    float* out = (float*)d_out;

    const long long* src = eidx;
    const long long* dst = eidx + NEDGES;

    // ---- workspace layout (h doubles as the per-layer scatter accumulator) ----
    char* p = (char*)d_ws;
    float*    h      = (float*)p;    p += (size_t)NODES * DIM * sizeof(float);   // 51.2 MB
    float*    gl     = (float*)p;    p += (size_t)NODES * DIM * sizeof(float);   // 51.2 MB
    float*    gr     = (float*)p;    p += (size_t)NODES * DIM * sizeof(float);   // 51.2 MB
    float*    logits = (float*)p;    p += (size_t)NEDGES * HEADS * sizeof(float);// 25.6 MB
    unsigned* maxv   = (unsigned*)p; p += (size_t)NODES * HEADS * sizeof(unsigned);
    float*    denom  = (float*)p;    p += (size_t)NODES * HEADS * sizeof(float);
    float*    stats  = (float*)p;    p += 2 * DIM * sizeof(float);

    const int gemm_blocks = (NODES / 16 + 7) / 8;             // 782 (8 waves/block)
    const int edge_blocks = (NEDGES + 7) / 8;                 // 200000
    const int init_blocks = (int)(((size_t)NODES * DIM + 255) / 256);
    const int node_blocks = (NODES + 127) / 128;              // 782

    const float* hin = x;
    for (int i = 0; i < NLAYERS; i++) {
        const float* Wl_i = Wl + (size_t)i * DIM * DIM;
        const float* Wr_i = Wr + (size_t)i * DIM * DIM;
        const float* We_i = We + (size_t)i * EDIM * DIM;
        const float* at_i = att + (size_t)i * DIM;            // HEADS*CH = 128 flat

        gemm_node_kernel<<<gemm_blocks, 256, 0, stream>>>(hin, Wl_i, bl + i * DIM, gl, NODES);
        gemm_node_kernel<<<gemm_blocks, 256, 0, stream>>>(hin, Wr_i, br + i * DIM, gr, NODES);
        init_kernel<<<init_blocks, 256, 0, stream>>>(h, denom, maxv, stats, NODES);
        edge_pass1_kernel<<<edge_blocks, 256, 0, stream>>>(gl, gr, ea, We_i, at_i,
                                                           src, dst, logits, maxv, NEDGES);
        edge_pass2_kernel<<<edge_blocks, 256, 0, stream>>>(gl, logits, maxv,
                                                           src, dst, denom, h, NEDGES);
        finalize_kernel<<<node_blocks, 128, 0, stream>>>(h, denom, bias + i * DIM, stats, NODES);
        bn_apply_kernel<<<node_blocks, 128, 0, stream>>>(h, stats, gamma + i * DIM,
                                                         beta + i * DIM, NODES);
        hin = h;
    }
    // final projection
    gemm_node_kernel<<<gemm_blocks, 256, 0, stream>>>(h, Wf, bf, out, NODES);
}